// get_model_44126493999705
// MI455X (gfx1250) — compile-verified
//
#include <hip/hip_runtime.h>
#include <math.h>

// ============================================================================
// PointNet++ MSG network for MI455X (gfx1250).
//
// Roofline: ~65 GFLOP of matmul, <1 GB traffic, every layer L2-resident
// (192 MB L2). Latency dominated by the sequential FPS scan and the BN-forced
// kernel chain, so fp32 WMMA (V_WMMA_F32_16X16X4_F32) is used: reference
// precision at negligible cost versus the serial-dependency floor.
//
// GEMM inner loop (after this round's cleanup):
//   ds_load_b64 (A frag from LDS) + global_load_b64 (B frag from padded W)
//   + v_wmma_f32_16x16x4_f32, fully branch-free; two Cout tiles per wave on
//   the wide layers for back-to-back independent WMMA issue.
// ============================================================================

typedef float v2f __attribute__((ext_vector_type(2)));
typedef float v8f __attribute__((ext_vector_type(8)));

#if __has_builtin(__builtin_amdgcn_wmma_f32_16x16x4_f32)
#define HAVE_WMMA_F32X4 1
#else
#define HAVE_WMMA_F32X4 0
#endif

#define NBATCH 8
#define NPTS0 2048
#define FPS_MAXN 2048

// ---------------------------------------------------------------------------
// Split (B,30,N) -> coords (B,N,3), feats (B,N,27)
// ---------------------------------------------------------------------------
__global__ void split_kernel(const float* __restrict__ x,
                             float* __restrict__ coords,
                             float* __restrict__ feats) {
  int i = blockIdx.x * blockDim.x + threadIdx.x;
  int total = NBATCH * NPTS0;
  if (i >= total) return;
  int b = i / NPTS0, n = i % NPTS0;
  const float* xb = x + (size_t)b * 30 * NPTS0;
  for (int c = 0; c < 3; ++c)  coords[(size_t)i * 3 + c]  = xb[(size_t)c * NPTS0 + n];
  for (int c = 0; c < 27; ++c) feats[(size_t)i * 27 + c] = xb[(size_t)(3 + c) * NPTS0 + n];
}

// ---------------------------------------------------------------------------
// Zero-pad weights: W (Cout x Cin) -> Wp (Cout x CinPad)
// ---------------------------------------------------------------------------
__global__ void padw_kernel(const float* __restrict__ W, float* __restrict__ Wp,
                            int Cin, int CinPad, int total) {
  int e = blockIdx.x * blockDim.x + threadIdx.x;
  if (e >= total) return;
  int c = e % CinPad, r = e / CinPad;
  Wp[e] = (c < Cin) ? W[(size_t)r * Cin + c] : 0.f;
}

// ---------------------------------------------------------------------------
// Farthest point sampling: one workgroup per batch, all state in LDS.
// Matches jax scan semantics: idx[0]=0; idx[i]=argmax after update with idx[i-1].
// ---------------------------------------------------------------------------
__global__ void __launch_bounds__(256)
fps_kernel(const float* __restrict__ xyz, int n, int S, int* __restrict__ out) {
  __shared__ float sx[FPS_MAXN], sy[FPS_MAXN], sz[FPS_MAXN], dist[FPS_MAXN];
  __shared__ float wval[8];
  __shared__ int   widx[8];
  __shared__ int   curIdx;
  int b = blockIdx.x, t = threadIdx.x;
  const float* base = xyz + (size_t)b * n * 3;
  for (int p = t; p < n; p += 256) {
    sx[p] = base[p * 3 + 0]; sy[p] = base[p * 3 + 1]; sz[p] = base[p * 3 + 2];
    dist[p] = 1e10f;
  }
  if (t == 0) { curIdx = 0; out[b * S + 0] = 0; }
  __syncthreads();
  for (int it = 1; it < S; ++it) {
    int cur = curIdx;
    float cx = sx[cur], cy = sy[cur], cz = sz[cur];
    float bestV = -1.f; int bestI = 0x7fffffff;
    for (int p = t; p < n; p += 256) {
      float dx = sx[p] - cx, dy = sy[p] - cy, dz = sz[p] - cz;
      float d = dx * dx + dy * dy + dz * dz;
      float dd = fminf(dist[p], d);
      dist[p] = dd;
      if (dd > bestV || (dd == bestV && p < bestI)) { bestV = dd; bestI = p; }
    }
    // wave32 argmax reduction (lowest index wins ties, as jnp.argmax)
    for (int off = 16; off > 0; off >>= 1) {
      float ov = __shfl_down(bestV, off);
      int   oi = __shfl_down(bestI, off);
      if (ov > bestV || (ov == bestV && oi < bestI)) { bestV = ov; bestI = oi; }
    }
    int wid = t >> 5;
    if ((t & 31) == 0) { wval[wid] = bestV; widx[wid] = bestI; }
    __syncthreads();
    if (t == 0) {
      float v = wval[0]; int i0 = widx[0];
      for (int w = 1; w < 8; ++w)
        if (wval[w] > v || (wval[w] == v && widx[w] < i0)) { v = wval[w]; i0 = widx[w]; }
      curIdx = i0; out[b * S + it] = i0;
    }
    __syncthreads();
  }
}

// ---------------------------------------------------------------------------
// new_xyz[b,s,:] = coords[b, fpsidx[b,s], :]
// ---------------------------------------------------------------------------
__global__ void gather_xyz_kernel(const float* __restrict__ xyz,
                                  const int* __restrict__ idx,
                                  float* __restrict__ out, int n, int S, int total) {
  int i = blockIdx.x * blockDim.x + threadIdx.x;
  if (i >= total) return;
  int b = i / S;
  int id = idx[i];
  for (int c = 0; c < 3; ++c)
    out[(size_t)i * 3 + c] = xyz[((size_t)b * n + id) * 3 + c];
}

// ---------------------------------------------------------------------------
// Ball query: one wave per centroid; first-K-in-index-order via ballot compaction.
// ---------------------------------------------------------------------------
__global__ void __launch_bounds__(32)
ballquery_kernel(const float* __restrict__ xyz, const float* __restrict__ new_xyz,
                 int n, int S, int K, float r2, int* __restrict__ out) {
  int g = blockIdx.x;          // b*S + s
  int b = g / S;
  int lane = threadIdx.x;
  float qx = new_xyz[(size_t)g * 3 + 0];
  float qy = new_xyz[(size_t)g * 3 + 1];
  float qz = new_xyz[(size_t)g * 3 + 2];
  const float* base = xyz + (size_t)b * n * 3;
  int cnt = 0, firstIdx = -1;
  for (int bp = 0; bp < n && cnt < K; bp += 32) {
    int p = bp + lane;
    bool ok = false;
    if (p < n) {
      float dx = base[p * 3 + 0] - qx, dy = base[p * 3 + 1] - qy, dz = base[p * 3 + 2] - qz;
      ok = (dx * dx + dy * dy + dz * dz) <= r2;
    }
    unsigned mm = (unsigned)__ballot(ok);
    if (firstIdx < 0 && mm) firstIdx = bp + __ffs(mm) - 1;
    if (ok) {
      int pos = cnt + __popc(mm & ((1u << lane) - 1u));
      if (pos < K) out[(size_t)g * K + pos] = p;
    }
    cnt += __popc(mm);
  }
  if (cnt < K) {
    int pad = (firstIdx >= 0) ? firstIdx : (n - 1);
    for (int pos = cnt + lane; pos < K; pos += 32) out[(size_t)g * K + pos] = pad;
  }
}

// ---------------------------------------------------------------------------
// Group: X[(b,s,k), 0..Cf-1] = feat[b, idx, :];  X[..., Cf..Cf+2] = xyz[idx]-new_xyz
// ---------------------------------------------------------------------------
__global__ void group_kernel(const float* __restrict__ feat, int Cf,
                             const float* __restrict__ xyz,
                             const float* __restrict__ new_xyz,
                             const int* __restrict__ idx,
                             int S, int K, int n, int CinPad,
                             float* __restrict__ X, int total) {
  int e = blockIdx.x * blockDim.x + threadIdx.x;
  if (e >= total) return;
  int Cin = Cf + 3;
  int c = e % Cin;
  int r = e / Cin;           // (b*S+s)*K + k
  int g = r / K;
  int b = g / S;
  int id = idx[r];
  float v;
  if (c < Cf) v = feat[((size_t)b * n + id) * Cf + c];
  else {
    int cc = c - Cf;
    v = xyz[((size_t)b * n + id) * 3 + cc] - new_xyz[(size_t)g * 3 + cc];
  }
  X[(size_t)r * CinPad + c] = v;
}

// ---------------------------------------------------------------------------
// SA3 concat: X[(b,n), 0..2] = l2_xyz; X[(b,n), 3..642] = l2_pts
// ---------------------------------------------------------------------------
__global__ void concat_sa3_kernel(const float* __restrict__ l2xyz,
                                  const float* __restrict__ l2pts,
                                  float* __restrict__ X, int Cpts, int CinPad, int total) {
  int e = blockIdx.x * blockDim.x + threadIdx.x;
  if (e >= total) return;
  int Cin = 3 + Cpts;
  int c = e % Cin;
  int r = e / Cin;
  X[(size_t)r * CinPad + c] = (c < 3) ? l2xyz[(size_t)r * 3 + c]
                                      : l2pts[(size_t)r * Cpts + (c - 3)];
}

// ---------------------------------------------------------------------------
// GEMM: Y[p,d] = sum_c A[p,c] * Wp[d,c] + bias[d], Wp zero-padded to CinPad,
// with optional fused BN+ReLU of the *input* during LDS staging.
// 16-row point tile in LDS; each wave owns one or two 16-wide Cout tiles.
// A frag (16x4 f32):  lane<16 -> K=0,1 ; lane>=16 -> K=2,3 ; M = lane&15
// B frag (4x16 f32):  same K split  ; N = lane&15
// D frag (16x16 f32): vgpr i -> row i + (lane<16?0:8), col = lane&15
// ---------------------------------------------------------------------------
__global__ void __launch_bounds__(128)
gemm_bn_kernel(const float* __restrict__ X, int ldx, int Cin, int CinPad, int lstr,
               const float* __restrict__ Wp, const float* __restrict__ bias,
               float* __restrict__ Y, int Cout,
               const float* __restrict__ mean, const float* __restrict__ rstd,
               const float* __restrict__ gsc, const float* __restrict__ bsh,
               int fuseBn, int tilesPerWave) {
  extern __shared__ float lA[];          // 16 x lstr, zero-padded cols >= Cin
  int p0 = blockIdx.x * 16;
  int t = threadIdx.x;
  for (int e = t; e < 16 * lstr; e += 128) {
    int row = e / lstr, col = e % lstr;
    float v = 0.f;
    if (col < Cin) {
      v = X[(size_t)(p0 + row) * ldx + col];
      if (fuseBn) v = fmaxf((v - mean[col]) * rstd[col] * gsc[col] + bsh[col], 0.f);
    }
    lA[e] = v;
  }
  __syncthreads();

  int wid = t >> 5, lane = t & 31;
  int m = lane & 15;
  int kb = (lane < 16) ? 0 : 2;          // this half-wave's K offset
  int nTiles = Cout >> 4;
  int t0 = blockIdx.y * 4 * tilesPerWave + wid;
  if (t0 >= nTiles) return;
  bool two = (tilesPerWave == 2) && (t0 + 4 < nTiles);
  int d0 = t0 * 16;
  int d1 = (t0 + 4) * 16;

  v8f acc0 = {}, acc1 = {};
  const float* w0 = Wp + (size_t)(d0 + m) * CinPad;   // B frag column n == m
  const float* w1 = Wp + (size_t)(two ? (d1 + m) : (d0 + m)) * CinPad;
  const float* arow = &lA[m * lstr];

  if (two) {
    for (int c0 = 0; c0 < CinPad; c0 += 4) {
      int ck = c0 + kb;                                 // even -> 8B aligned
      v2f a  = *(const v2f*)(arow + ck);
      v2f b0 = *(const v2f*)(w0 + ck);
      v2f b1 = *(const v2f*)(w1 + ck);
#if HAVE_WMMA_F32X4
      acc0 = __builtin_amdgcn_wmma_f32_16x16x4_f32(false, a, false, b0,
                                                   (short)0, acc0, false, false);
      acc1 = __builtin_amdgcn_wmma_f32_16x16x4_f32(false, a, false, b1,
                                                   (short)0, acc1, false, false);
#else
      int rb = (lane < 16) ? 0 : 8;
      for (int k = 0; k < 4; ++k) {
        float aw = lA[(rb + 0) * lstr + c0 + k]; (void)aw;
#pragma unroll
        for (int i = 0; i < 8; ++i) {
          float av = lA[(rb + i) * lstr + c0 + k];
          acc0[i] += av * Wp[(size_t)(d0 + m) * CinPad + c0 + k];
          acc1[i] += av * Wp[(size_t)(d1 + m) * CinPad + c0 + k];
        }
      }
#endif
    }
  } else {
    for (int c0 = 0; c0 < CinPad; c0 += 4) {
      int ck = c0 + kb;
      v2f a  = *(const v2f*)(arow + ck);
      v2f b0 = *(const v2f*)(w0 + ck);
#if HAVE_WMMA_F32X4
      acc0 = __builtin_amdgcn_wmma_f32_16x16x4_f32(false, a, false, b0,
                                                   (short)0, acc0, false, false);
#else
      int rb = (lane < 16) ? 0 : 8;
      for (int k = 0; k < 4; ++k)
#pragma unroll
        for (int i = 0; i < 8; ++i)
          acc0[i] += lA[(rb + i) * lstr + c0 + k] *
                     Wp[(size_t)(d0 + m) * CinPad + c0 + k];
#endif
    }
  }

  int rbase = (lane < 16) ? 0 : 8;
  float bc0 = bias[d0 + m];
#pragma unroll
  for (int i = 0; i < 8; ++i)
    Y[(size_t)(p0 + rbase + i) * Cout + d0 + m] = acc0[i] + bc0;
  if (two) {
    float bc1 = bias[d1 + m];
#pragma unroll
    for (int i = 0; i < 8; ++i)
      Y[(size_t)(p0 + rbase + i) * Cout + d1 + m] = acc1[i] + bc1;
  }
}

// ---------------------------------------------------------------------------
// Per-channel biased mean/var over Npts rows -> mean, rstd=1/sqrt(var+eps)
// ---------------------------------------------------------------------------
__global__ void __launch_bounds__(256)
stats_kernel(const float* __restrict__ Y, int Npts, int C,
             float* __restrict__ mean, float* __restrict__ rstd) {
  __shared__ float ss[256], ss2[256];
  int d = blockIdx.x, t = threadIdx.x;
  float s = 0.f, s2 = 0.f;
  for (int r = t; r < Npts; r += 256) {
    float v = Y[(size_t)r * C + d];
    s += v; s2 += v * v;
  }
  ss[t] = s; ss2[t] = s2; __syncthreads();
  for (int off = 128; off > 0; off >>= 1) {
    if (t < off) { ss[t] += ss[t + off]; ss2[t] += ss2[t + off]; }
    __syncthreads();
  }
  if (t == 0) {
    float mu = ss[0] / (float)Npts;
    float var = ss2[0] / (float)Npts - mu * mu;
    mean[d] = mu;
    rstd[d] = rsqrtf(var + 1e-5f);
  }
}

// ---------------------------------------------------------------------------
// Fused BN+ReLU+max-over-K, writing into the concat output slice.
// ---------------------------------------------------------------------------
__global__ void __launch_bounds__(256)
maxpool_bn_kernel(const float* __restrict__ Y, int K, int C,
                  const float* __restrict__ mean, const float* __restrict__ rstd,
                  const float* __restrict__ gsc, const float* __restrict__ bsh,
                  float* __restrict__ out, int Ctot, int co) {
  int grp = blockIdx.x, t = threadIdx.x;
  for (int c = t; c < C; c += 256) {
    float mu = mean[c], rs = rstd[c], gg = gsc[c], bb = bsh[c];
    float mv = -1e30f;
    for (int k = 0; k < K; ++k) {
      float v = Y[((size_t)grp * K + k) * C + c];
      v = fmaxf((v - mu) * rs * gg + bb, 0.f);
      mv = fmaxf(mv, v);
    }
    out[(size_t)grp * Ctot + co + c] = mv;
  }
}

// ---------------------------------------------------------------------------
// Final: out[b] = sigmoid( b3 + sum_c relu(bn(yfc2[b,c])) * W3[c] ), C=256
// ---------------------------------------------------------------------------
__global__ void __launch_bounds__(256)
fc3_sigmoid_kernel(const float* __restrict__ Y,
                   const float* __restrict__ mean, const float* __restrict__ rstd,
                   const float* __restrict__ gsc, const float* __restrict__ bsh,
                   const float* __restrict__ W3, const float* __restrict__ b3,
                   float* __restrict__ out) {
  __shared__ float ss[256];
  int b = blockIdx.x, t = threadIdx.x;
  float v = Y[(size_t)b * 256 + t];
  v = fmaxf((v - mean[t]) * rstd[t] * gsc[t] + bsh[t], 0.f);
  ss[t] = v * W3[t];
  __syncthreads();
  for (int off = 128; off > 0; off >>= 1) {
    if (t < off) ss[t] += ss[t + off];
    __syncthreads();
  }
  if (t == 0) out[b] = 1.f / (1.f + expf(-(ss[0] + b3[0])));
}

// ===========================================================================
// Host side
// ===========================================================================
struct LayerP { const float *W, *b, *g, *be; int cin, cout; };

static void launch_gemm(const float* X, int ldx, int Cin, int CinPad,
                        const float* Wp, const LayerP& L, float* Y, int NptsPad,
                        const float* mn, const float* rs,
                        const float* g, const float* be, int fuse,
                        hipStream_t s) {
  int lstr = ((size_t)16 * (CinPad + 4) * 4 <= 65536) ? (CinPad + 4) : CinPad;
  size_t sh = (size_t)16 * lstr * 4;
  int nTiles = L.cout / 16;
  int tpw = (nTiles % 8 == 0) ? 2 : 1;
  dim3 grid(NptsPad / 16, (nTiles + 4 * tpw - 1) / (4 * tpw));
  gemm_bn_kernel<<<grid, 128, sh, s>>>(X, ldx, Cin, CinPad, lstr,
                                       Wp, L.b, Y, L.cout, mn, rs, g, be, fuse, tpw);
}

// Run an MLP chain (GEMM + stats per layer, BN+ReLU fused into next GEMM's
// staging). Returns the raw (pre-BN) activations of the last layer; mn/rs
// hold the last layer's statistics afterwards.
static const float* run_chain(const float* X0, int ldx0, int Cin0, int CinPad0,
                              const LayerP* L, int nL, int Npts, int NptsPad,
                              float* bufA, float* bufB, float* wpad,
                              float* mn, float* rs, hipStream_t s) {
  const float* cur = X0;
  int ldx = ldx0, Cin = Cin0, CinPad = CinPad0;
  const float *bg = nullptr, *bb = nullptr;
  int fuse = 0;
  const float* out = nullptr;
  for (int i = 0; i < nL; ++i) {
    const float* Wp = L[i].W;
    if (CinPad != Cin) {                 // ragged first layer: zero-pad weights
      int total = L[i].cout * CinPad;
      padw_kernel<<<(total + 255) / 256, 256, 0, s>>>(L[i].W, wpad, Cin, CinPad, total);
      Wp = wpad;
    }
    float* O = (i & 1) ? bufB : bufA;
    launch_gemm(cur, ldx, Cin, CinPad, Wp, L[i], O, NptsPad, mn, rs, bg, bb, fuse, s);
    stats_kernel<<<L[i].cout, 256, 0, s>>>(O, Npts, L[i].cout, mn, rs);
    cur = O; ldx = L[i].cout; Cin = L[i].cout; CinPad = L[i].cout;
    fuse = 1; bg = L[i].g; bb = L[i].be;
    out = O;
  }
  return out;
}

static char* wsTake(char*& p, size_t bytes) {
  char* r = p;
  p += (bytes + 255) & ~(size_t)255;
  return r;
}

extern "C" void kernel_launch(void* const* d_in, const int* in_sizes, int n_in,
                              void* d_out, int out_size, void* d_ws, size_t ws_size,
                              hipStream_t stream) {
  (void)in_sizes; (void)n_in; (void)out_size; (void)ws_size;
  // ---- input unpacking (setup_inputs insertion order) ----
  int ci = 0;
  const float* x_in = (const float*)d_in[ci++];  // xyz (B,30,N)
  LayerP sa1[3][3], sa2[3][3], sa3L[3], fc1L, fc2L;
  const float *fc3W, *fc3b;
  auto rdL = [&](int cin, int cout, bool bn) -> LayerP {
    LayerP L;
    L.W = (const float*)d_in[ci++];
    L.b = (const float*)d_in[ci++];
    if (bn) { L.g = (const float*)d_in[ci++]; L.be = (const float*)d_in[ci++]; }
    else    { L.g = nullptr; L.be = nullptr; }
    L.cin = cin; L.cout = cout;
    return L;
  };
  const int sa1w[3][3] = {{32, 32, 64}, {64, 64, 128}, {64, 96, 128}};
  const int sa2w[3][3] = {{64, 64, 128}, {128, 128, 256}, {128, 128, 256}};
  const int sa3w[3] = {256, 512, 1024};
  for (int br = 0; br < 3; ++br) { int c = 30;  for (int l = 0; l < 3; ++l) { sa1[br][l] = rdL(c, sa1w[br][l], true); c = sa1w[br][l]; } }
  for (int br = 0; br < 3; ++br) { int c = 323; for (int l = 0; l < 3; ++l) { sa2[br][l] = rdL(c, sa2w[br][l], true); c = sa2w[br][l]; } }
  { int c = 643; for (int l = 0; l < 3; ++l) { sa3L[l] = rdL(c, sa3w[l], true); c = sa3w[l]; } }
  fc1L = rdL(1024, 512, true);
  fc2L = rdL(512, 256, true);
  fc3W = (const float*)d_in[ci++];
  fc3b = (const float*)d_in[ci++];

  // ---- workspace carve-up ----
  char* wp = (char*)d_ws;
  float* coords  = (float*)wsTake(wp, (size_t)NBATCH * NPTS0 * 3 * 4);
  float* feats   = (float*)wsTake(wp, (size_t)NBATCH * NPTS0 * 27 * 4);
  int*   fpsidx  = (int*)  wsTake(wp, (size_t)NBATCH * 512 * 4);
  float* newxyz1 = (float*)wsTake(wp, (size_t)NBATCH * 512 * 3 * 4);
  float* newxyz2 = (float*)wsTake(wp, (size_t)NBATCH * 128 * 3 * 4);
  float* l1pts   = (float*)wsTake(wp, (size_t)NBATCH * 512 * 320 * 4);
  float* l2pts   = (float*)wsTake(wp, (size_t)NBATCH * 128 * 640 * 4);
  float* meanB   = (float*)wsTake(wp, 1024 * 4);
  float* rstdB   = (float*)wsTake(wp, 1024 * 4);
  float* wpadB   = (float*)wsTake(wp, (size_t)256 * 644 * 4);      // padded-W scratch
  int*   idxbuf  = (int*)  wsTake(wp, (size_t)NBATCH * 512 * 128 * 4);
  float* xfc     = (float*)wsTake(wp, (size_t)16 * 1024 * 4);
  float* yfc1    = (float*)wsTake(wp, (size_t)16 * 512 * 4);
  float* yfc2    = (float*)wsTake(wp, (size_t)16 * 256 * 4);
  float* Xg      = (float*)wsTake(wp, (size_t)131072 * 324 * 4);   // max grouped tile
  float* bufA    = (float*)wsTake(wp, (size_t)524288 * 128 * 4);   // activation ping
  float* bufB    = (float*)wsTake(wp, (size_t)524288 * 128 * 4);   // activation pong

  // ---- split input ----
  {
    int total = NBATCH * NPTS0;
    split_kernel<<<(total + 255) / 256, 256, 0, stream>>>(x_in, coords, feats);
  }

  // ---- generic SA-MSG level ----
  auto run_sa_msg = [&](const float* cds, int n, const float* ft, int Cf, int S,
                        const float* radii, const int* Ks, LayerP (*L)[3],
                        float* newxyz, float* ptsOut, int Ctot) {
    fps_kernel<<<NBATCH, 256, 0, stream>>>(cds, n, S, fpsidx);
    int tot = NBATCH * S;
    gather_xyz_kernel<<<(tot + 255) / 256, 256, 0, stream>>>(cds, fpsidx, newxyz, n, S, tot);
    int co = 0;
    for (int br = 0; br < 3; ++br) {
      int K = Ks[br];
      float r2 = radii[br] * radii[br];
      ballquery_kernel<<<NBATCH * S, 32, 0, stream>>>(cds, newxyz, n, S, K, r2, idxbuf);
      int Cin = Cf + 3, CinPad = (Cin + 3) & ~3;
      int Npts = NBATCH * S * K;                // multiple of 16 for all branches
      int total = Npts * Cin;
      group_kernel<<<(total + 255) / 256, 256, 0, stream>>>(
          ft, Cf, cds, newxyz, idxbuf, S, K, n, CinPad, Xg, total);
      const float* last = run_chain(Xg, CinPad, Cin, CinPad, L[br], 3,
                                    Npts, Npts, bufA, bufB, wpadB, meanB, rstdB, stream);
      const LayerP& LL = L[br][2];
      maxpool_bn_kernel<<<NBATCH * S, 256, 0, stream>>>(
          last, K, LL.cout, meanB, rstdB, LL.g, LL.be, ptsOut, Ctot, co);
      co += LL.cout;
    }
  };

  const float r1[3] = {0.1f, 0.2f, 0.4f};
  const int   k1[3] = {16, 32, 128};
  run_sa_msg(coords, NPTS0, feats, 27, 512, r1, k1, sa1, newxyz1, l1pts, 320);

  const float r2a[3] = {0.2f, 0.4f, 0.8f};
  const int   k2[3] = {32, 64, 128};
  run_sa_msg(newxyz1, 512, l1pts, 320, 128, r2a, k2, sa2, newxyz2, l2pts, 640);

  // ---- SA3 (group-all): 1024 points x 643 channels ----
  {
    int rows = NBATCH * 128, Cin = 643, CinPad = 644;
    int total = rows * Cin;
    concat_sa3_kernel<<<(total + 255) / 256, 256, 0, stream>>>(
        newxyz2, l2pts, Xg, 640, CinPad, total);
    const float* last = run_chain(Xg, CinPad, Cin, CinPad, sa3L, 3,
                                  rows, rows, bufA, bufB, wpadB, meanB, rstdB, stream);
    // max over the 128 points per batch -> xfc (B,1024), stride 1024
    maxpool_bn_kernel<<<NBATCH, 256, 0, stream>>>(
        last, 128, 1024, meanB, rstdB, sa3L[2].g, sa3L[2].be, xfc, 1024, 0);
  }

  // ---- FC head (Npts=8 valid rows, padded to 16 for the WMMA tile) ----
  launch_gemm(xfc, 1024, 1024, 1024, fc1L.W, fc1L, yfc1, 16,
              meanB, rstdB, nullptr, nullptr, 0, stream);
  stats_kernel<<<512, 256, 0, stream>>>(yfc1, NBATCH, 512, meanB, rstdB);
  launch_gemm(yfc1, 512, 512, 512, fc2L.W, fc2L, yfc2, 16,
              meanB, rstdB, fc1L.g, fc1L.be, 1, stream);
  stats_kernel<<<256, 256, 0, stream>>>(yfc2, NBATCH, 256, meanB, rstdB);
  fc3_sigmoid_kernel<<<NBATCH, 256, 0, stream>>>(
      yfc2, meanB, rstdB, fc2L.g, fc2L.be, fc3W, fc3b, (float*)d_out);
}